// Postprocess_33200097198982
// MI455X (gfx1250) — compile-verified
//
#include <hip/hip_runtime.h>
#include <stdint.h>

#define RR 4
#define KW 9
#define PATCH (KW * KW * KW)   // 729
#define NCHUNK ((PATCH + 3) / 4)   // 183
#define KPAD (NCHUNK * 4)          // 732
#define THRESH 4.0f
#define MAX_DET 4096
#define DD 64
#define HH 512
#define WW 512
#define HW (HH * WW)
#define CAND_CAP 8192
#define NSORT 8192
#define PSX 100.0f
#define PSY 100.0f
#define PSZ 20.0f
#define ZMIN -400.0f

typedef float v8f __attribute__((ext_vector_type(8)));
typedef float v2f __attribute__((ext_vector_type(2)));

struct Cand { float v; uint32_t i; };

__global__ void k_init(uint32_t* counter) { *counter = 0u; }

// Pass 1: threshold + separable 9x9 max over X,Y per z-slice.
// 32x32 output tile, 40x40 halo in LDS. OOB halo = 0 (valid: pt >= 0 always,
// so 0-padding never changes a window max that matters for mp > 0).
__global__ __launch_bounds__(256) void k_xymax(const float* __restrict__ pred,
                                               float* __restrict__ mxy) {
    __shared__ float tin[40][40];
    __shared__ float xm[40][32];
    const int z = blockIdx.z;
    const int by = blockIdx.y * 32, bx = blockIdx.x * 32;
    const int tid = threadIdx.x;
    const float* slice = pred + (size_t)z * HW;

    for (int i = tid; i < 40 * 40; i += 256) {
        int ly = i / 40, lx = i % 40;
        int gy = by + ly - RR, gx = bx + lx - RR;
        float v = 0.0f;
        if ((unsigned)gy < HH && (unsigned)gx < WW) {
            float p = slice[gy * WW + gx];
            v = p > THRESH ? p : 0.0f;
        }
        tin[ly][lx] = v;
    }
    __syncthreads();
    for (int i = tid; i < 40 * 32; i += 256) {
        int ly = i / 32, lx = i % 32;
        float m = tin[ly][lx];
#pragma unroll
        for (int t = 1; t < KW; ++t) m = fmaxf(m, tin[ly][lx + t]);
        xm[ly][lx] = m;
    }
    __syncthreads();
    for (int i = tid; i < 32 * 32; i += 256) {
        int ly = i / 32, lx = i % 32;
        float m = xm[ly][lx];
#pragma unroll
        for (int t = 1; t < KW; ++t) m = fmaxf(m, xm[ly + t][lx]);
        mxy[(size_t)z * HW + (size_t)(by + ly) * WW + (bx + lx)] = m;
    }
}

// Pass 2: per-(y,x)-column sliding 9-window max over Z + peak test + compaction.
// Consecutive threads = consecutive x -> coalesced per-plane loads.
// global_prefetch 8 planes ahead hides HBM latency in this dependent loop.
__global__ __launch_bounds__(256) void k_zscan(const float* __restrict__ pred,
                                               const float* __restrict__ mxy,
                                               uint32_t* __restrict__ counter,
                                               Cand* __restrict__ cands) {
    const int col = blockIdx.x * 256 + threadIdx.x;   // 0..HW-1
    float w[KW];
#pragma unroll
    for (int j = 0; j < KW; ++j) {
        int zz = j - RR;
        w[j] = (zz >= 0 && zz < DD) ? mxy[(size_t)zz * HW + col] : 0.0f;
    }
    for (int z = 0; z < DD; ++z) {
        if (z + 8 < DD)
            __builtin_prefetch(&pred[(size_t)(z + 8) * HW + col], 0, 0);
        if (z + RR + 1 + 8 < DD)
            __builtin_prefetch(&mxy[(size_t)(z + RR + 1 + 8) * HW + col], 0, 0);
        float mp = w[0];
#pragma unroll
        for (int j = 1; j < KW; ++j) mp = fmaxf(mp, w[j]);
        float p = pred[(size_t)z * HW + col];
        float pt = p > THRESH ? p : 0.0f;
        if (mp > 0.0f && mp == pt) {
            uint32_t slot = atomicAdd(counter, 1u);
            if (slot < CAND_CAP) {
                cands[slot].v = pt;
                cands[slot].i = (uint32_t)(z * HW + col);
            }
        }
#pragma unroll
        for (int j = 0; j < KW - 1; ++j) w[j] = w[j + 1];
        int zn = z + RR + 1;
        w[KW - 1] = (zn < DD) ? mxy[(size_t)zn * HW + col] : 0.0f;
    }
}

// Pass 3: single-block bitonic sort, descending on key = val_bits<<32 | ~index.
// All candidate values are > 4.0 (positive floats monotone as uint), so this
// reproduces jax top_k order exactly: value desc, index asc. Pad key = 0.
__global__ __launch_bounds__(1024) void k_sort(const uint32_t* __restrict__ counter,
                                               const Cand* __restrict__ cands,
                                               float* __restrict__ pvals,
                                               uint32_t* __restrict__ pidx) {
    __shared__ uint64_t key[NSORT];
    const int tid = threadIdx.x;
    uint32_t cnt = *counter;
    if (cnt > CAND_CAP) cnt = CAND_CAP;
    for (int i = tid; i < NSORT; i += 1024) {
        uint64_t k = 0;
        if (i < (int)cnt) {
            uint32_t vb = __float_as_uint(cands[i].v);
            k = ((uint64_t)vb << 32) | (uint64_t)(uint32_t)(~cands[i].i);
        }
        key[i] = k;
    }
    __syncthreads();
    for (int k = 2; k <= NSORT; k <<= 1) {
        for (int j = k >> 1; j > 0; j >>= 1) {
            for (int t = tid; t < NSORT; t += 1024) {
                int ixj = t ^ j;
                if (ixj > t) {
                    uint64_t a = key[t], b = key[ixj];
                    bool desc = ((t & k) == 0);
                    if (desc ? (a < b) : (a > b)) { key[t] = b; key[ixj] = a; }
                }
            }
            __syncthreads();
        }
    }
    for (int i = tid; i < MAX_DET; i += 1024) {
        uint64_t k = key[i];
        pvals[i] = k ? __uint_as_float((uint32_t)(k >> 32)) : 0.0f;
        pidx[i]  = k ? (uint32_t)~(uint32_t)k : 0u;
    }
}

// Pass 4: centroid moments via V_WMMA_F32_16X16X4_F32.
// One wave handles 16 peaks (M). N=16 columns, only n<4 used:
//   B[k][n] = {1, dx-4, dy-4, dz-4}[n] for k<729, else 0.
// K=729 accumulated in 183 chunks of 4.
// ISA layouts (wave32): lane l -> A row M=l%16, K = v + 2*(l/16) in VGPR v;
// B column N=l%16 with same K striping; C: lane l VGPR v holds
// C[v + 8*(l/16)][l%16].
//
// Per-block LDS LUTs kill the per-iteration div/mod and the exec-mask
// branch chains: lut[k] = {pack(dz,dy,dx), goff}, bfilt[5][...] holds the
// four filter rows + a zero row for lanes with column >= 4. Guarded loads
// are branchless: cndmask the address to 0 (always-safe) and the value to 0.
__global__ __launch_bounds__(256) void k_centroid(const float* __restrict__ pred,
                                                  const float* __restrict__ pvals,
                                                  const uint32_t* __restrict__ pidx,
                                                  float* __restrict__ out) {
    __shared__ alignas(16) int2 lut[KPAD + 4];       // {pack, goff}
    __shared__ alignas(16) float bfilt[5][KPAD + 4]; // rows: 1, xf, yf, zf, 0
    __shared__ float cs[8][16][16];
    const int tid = threadIdx.x;

    for (int i = tid; i < KPAD + 4; i += 256) {
        int pack = 0x7F7F7F;   // forces out-of-range -> contributes 0
        int goff = 0;
        float b0 = 0.f, b1 = 0.f, b2 = 0.f, b3 = 0.f;
        if (i < PATCH) {
            int dz = i / 81;
            int r  = i - dz * 81;
            int dy = r / 9;
            int dx = r - dy * 9;
            pack = dz | (dy << 8) | (dx << 16);
            goff = dz * HW + dy * WW + dx;
            b0 = 1.0f;
            b1 = (float)(dx - RR);
            b2 = (float)(dy - RR);
            b3 = (float)(dz - RR);
        }
        lut[i] = make_int2(pack, goff);
        bfilt[0][i] = b0; bfilt[1][i] = b1; bfilt[2][i] = b2; bfilt[3][i] = b3;
        bfilt[4][i] = 0.0f;
    }
    __syncthreads();

    const int wv = tid >> 5, lane = tid & 31;
    const int mrow = lane & 15, half = lane >> 4;
    const int g = blockIdx.x * 8 + wv;     // wave group of 16 peaks
    const int pk = g * 16 + mrow;

    const uint32_t idx = pidx[pk];
    const int zb = (int)(idx / HW);
    const int rem = (int)(idx % HW);
    const int yb = rem / WW, xb = rem % WW;
    const int z0 = zb - RR, y0 = yb - RR, x0 = xb - RR;
    const int base = z0 * HW + y0 * WW + x0;      // may be negative; masked below
    const float* bf = bfilt[mrow < 4 ? mrow : 4];

    v8f acc = {};
    for (int c = 0; c < NCHUNK; ++c) {
        const int k = c * 4 + 2 * half;           // even -> 16B-aligned LUT pair
        const int4 pg = *(const int4*)&lut[k];    // {pack0,goff0,pack1,goff1}
        const v2f b = *(const v2f*)&bf[k];
        v2f a;
#pragma unroll
        for (int v = 0; v < 2; ++v) {
            const int pack = (v == 0) ? pg.x : pg.z;
            const int goff = (v == 0) ? pg.y : pg.w;
            const int az = z0 + (pack & 255);
            const int ay = y0 + ((pack >> 8) & 255);
            const int ax = x0 + (pack >> 16);
            const bool ok = ((unsigned)az < DD) & ((unsigned)ay < HH) &
                            ((unsigned)ax < WW);
            const int addr = ok ? (base + goff) : 0;   // always in-range
            const float ld = pred[addr];
            a[v] = ok ? ld : 0.0f;
        }
        acc = __builtin_amdgcn_wmma_f32_16x16x4_f32(
            /*neg_a=*/false, a, /*neg_b=*/false, b,
            /*c_mod=*/(short)0, acc, /*reuse_a=*/false, /*reuse_b=*/false);
    }
#pragma unroll
    for (int v = 0; v < 8; ++v) cs[wv][v + 8 * half][mrow] = acc[v];
    __syncthreads();

    if (lane < 16) {
        const int p2 = g * 16 + lane;
        const float val = pvals[p2];
        const uint32_t id2 = pidx[p2];
        const int z2 = (int)(id2 / HW);
        const int r2 = (int)(id2 % HW);
        const int y2 = r2 / WW, x2 = r2 % WW;
        const float s0 = cs[wv][lane][0];
        const float sx = cs[wv][lane][1];
        const float sy = cs[wv][lane][2];
        const float sz = cs[wv][lane][3];
        const float sd = s0 > 0.0f ? s0 : 1.0f;
        const bool valid = val > 0.0f;
        const float xr = ((float)x2 + sx / sd + 0.5f) * PSX;
        const float yr = ((float)y2 + sy / sd + 0.5f) * PSY;
        const float zr = ((float)z2 + sz / sd + 0.5f) * PSZ + ZMIN;
        out[p2 * 3 + 0]        = valid ? xr : 0.0f;
        out[p2 * 3 + 1]        = valid ? yr : 0.0f;
        out[p2 * 3 + 2]        = valid ? zr : 0.0f;
        out[MAX_DET * 3 + p2]  = valid ? val : 0.0f;
        out[MAX_DET * 4 + p2]  = valid ? 1.0f : 0.0f;
    }
}

extern "C" void kernel_launch(void* const* d_in, const int* in_sizes, int n_in,
                              void* d_out, int out_size, void* d_ws, size_t ws_size,
                              hipStream_t stream) {
    (void)in_sizes; (void)n_in; (void)out_size; (void)ws_size;
    const float* pred = (const float*)d_in[0];
    float* out = (float*)d_out;

    // Workspace layout: 64MB mxy + counter + candidates + sorted peak list.
    float* mxy = (float*)d_ws;
    uint8_t* p = (uint8_t*)d_ws + sizeof(float) * (size_t)DD * HW;
    uint32_t* counter = (uint32_t*)p;         p += 256;
    Cand* cands = (Cand*)p;                   p += sizeof(Cand) * CAND_CAP;
    float* pvals = (float*)p;                 p += sizeof(float) * MAX_DET;
    uint32_t* pidx = (uint32_t*)p;

    k_init<<<1, 1, 0, stream>>>(counter);
    dim3 g1(WW / 32, HH / 32, DD);
    k_xymax<<<g1, 256, 0, stream>>>(pred, mxy);
    k_zscan<<<HW / 256, 256, 0, stream>>>(pred, mxy, counter, cands);
    k_sort<<<1, 1024, 0, stream>>>(counter, cands, pvals, pidx);
    k_centroid<<<MAX_DET / 128, 256, 0, stream>>>(pred, pvals, pidx, out);
}